// TritonGatherConv_80221399155593
// MI455X (gfx1250) — compile-verified
//
#include <hip/hip_runtime.h>
#include <hip/hip_bf16.h>
#include <math.h>

// Problem constants (from the reference)
#define BB 2
#define LL 2048
#define CC 512
#define HH 8
#define DD 64
#define KK 64
#define HALF_S 16
#define SS 33
#define MAXF 16.0f
#define MINF 1.0f
#define MAX_RECEPTIVE 256.0f   // HALF_S * MAXF
#define MROWS (BB * LL)        // 4096

// GEMM tiling
#define BM 64                  // block tile M
#define BN 32                  // block tile N
#define BK 32                  // K chunk staged in LDS
#define LDS_STRIDE 33          // BK + 1 pad DWORD (inserted by TDM pad feature)

typedef __attribute__((ext_vector_type(2))) float v2f;
typedef __attribute__((ext_vector_type(8))) float v8f;
typedef __attribute__((ext_vector_type(4))) unsigned int v4u;
typedef __attribute__((ext_vector_type(8))) int v8i;
typedef __attribute__((ext_vector_type(4))) int v4i;

__device__ __forceinline__ float silu_f(float x) {
    return x / (1.0f + __expf(-x));
}
__device__ __forceinline__ float sigmoid_f(float x) {
    return 1.0f / (1.0f + __expf(-x));
}

// ---------------------------------------------------------------------------
// TDM: DMA a tileRows x 32 fp32 tile (row stride 512 floats) from global into
// LDS at byte offset ldsOff, inserting 1 pad DWORD after every 32 DWORDs so
// rows land at stride LDS_STRIDE floats (bank-conflict-free reads).
// D# layout per CDNA5 ISA.
// ---------------------------------------------------------------------------
#if __has_builtin(__builtin_amdgcn_tensor_load_to_lds)
#define HAVE_TDM 1
#else
#define HAVE_TDM 0
#endif

__device__ __forceinline__ void tdm_load_tile(const float* gbase,
                                              unsigned ldsOff,
                                              int tileRows) {
#if HAVE_TDM
    unsigned long long ga = (unsigned long long)(uintptr_t)gbase;
    v4u g0;
    g0.x = 1u;                                    // count=1 (valid user D#)
    g0.y = ldsOff;                                // lds_addr (bytes)
    g0.z = (unsigned)(ga & 0xffffffffu);          // global_addr[31:0]
    g0.w = (unsigned)((ga >> 32) & 0x01ffffffu)   // global_addr[56:32]
           | (2u << 30);                          // type=2 ("image")
    v8i g1;
    g1[0] = (int)((2u << 16)      // data_size = 4B
                | (1u << 20)      // pad_enable
                | (4u << 22)      // pad_interval code 4 -> every 32 DWORDs
                | (0u << 25));    // pad_amount  code 0 -> 1 DWORD
    g1[1] = (int)(512u << 16);    // tensor_dim0 = 512 (lo16 in bits 63:48)
    g1[2] = (int)(4096u << 16);   // tensor_dim0 hi = 0 ; tensor_dim1 lo16
    g1[3] = (int)(32u << 16);     // tensor_dim1 hi = 0 ; tile_dim0 = 32
    g1[4] = tileRows;             // tile_dim1 ; tile_dim2 = 0 (2D tile)
    g1[5] = 512;                  // tensor_dim0_stride (floats), lo32
    g1[6] = 0;                    // stride0 hi ; tensor_dim1_stride lo (unused)
    g1[7] = 0;
    v4i z4 = {};
#if defined(__clang_major__) && (__clang_major__ >= 23)
    v8i z8 = {};
    __builtin_amdgcn_tensor_load_to_lds(g0, g1, z4, z4, z8, 0);
#else
    __builtin_amdgcn_tensor_load_to_lds(g0, g1, z4, z4, 0);
#endif
#endif
}

// ---------------------------------------------------------------------------
// Kernel 1: wave = silu(x @ Ww^T + bw)  ->  freq, phase  [B*L, H]
// ---------------------------------------------------------------------------
__global__ void wave_kernel(const float* __restrict__ x,
                            const float* __restrict__ Ww,
                            const float* __restrict__ bw,
                            float* __restrict__ freq,
                            float* __restrict__ phase) {
    const int t   = threadIdx.x;
    const int row = blockIdx.x * 16 + (t >> 4);
    const int col = t & 15;
    const float* xr = x + (size_t)row * CC;
    const float* wr = Ww + (size_t)col * CC;
    float acc = bw[col];
#pragma unroll 8
    for (int c = 0; c < CC; c += 4) {
        float4 xv = *(const float4*)(xr + c);
        float4 wv = *(const float4*)(wr + c);
        acc = fmaf(xv.x, wv.x, acc);
        acc = fmaf(xv.y, wv.y, acc);
        acc = fmaf(xv.z, wv.z, acc);
        acc = fmaf(xv.w, wv.w, acc);
    }
    const float w = silu_f(acc);
    if (col < HH) {
        freq[(size_t)row * HH + col] = sigmoid_f(w) * (MAXF - MINF) + MINF;
    } else {
        phase[(size_t)row * HH + (col - HH)] = tanhf(w) * MAXF;
    }
}

// ---------------------------------------------------------------------------
// Kernel 2/4: out[m,n] = silu( sum_c A[m,c] * W[n,c] + bias[n] )
// TDM-fed, LDS double-buffered, fp32 WMMA (V_WMMA_F32_16X16X4_F32).
// Block: 256 threads = 8 waves as 4(M) x 2(N); block tile 64x32; K chunk 32.
// Fragment layouts per ISA 7.12.2 (wave32):
//   A 16x4: lane l -> row m0+(l&15), K pair k+2*(l>>4)   (float2)
//   B 4x16: lane l -> col n0+(l&15), K pair k+2*(l>>4)   (float2, B[c,n]=W[n,c])
//   D:      v[i] -> row i + 8*(l>>4), col l&15
// ---------------------------------------------------------------------------
__global__ void gemm_tdm_wmma_silu(const float* __restrict__ A,
                                   const float* __restrict__ W,
                                   const float* __restrict__ bias,
                                   float* __restrict__ out) {
    __shared__ float Ap[2][BM * LDS_STRIDE];   // 2 x 64 x 33 floats
    __shared__ float Bp[2][BN * LDS_STRIDE];   // 2 x 32 x 33 floats

    const int tid  = threadIdx.x;
    const int lane = tid & 31;
    const int wid  = tid >> 5;        // 0..7
    const int wm   = wid >> 1;        // 0..3  (16-row subtile)
    const int wn   = wid & 1;         // 0..1  (16-col subtile)
    const int half = lane >> 4;
    const int lr   = lane & 15;

    const int m0 = blockIdx.y * BM;   // gridDim.y = MROWS/BM = 64
    const int n0 = blockIdx.x * BN;   // gridDim.x = CC/BN   = 16

    const int NSTEP = CC / BK;        // 16

#if HAVE_TDM
    // Prime the pipeline: chunk 0 into buffer 0 (wave 0 only; TDM ignores EXEC,
    // issues once per wave).
    if (wid == 0) {
        tdm_load_tile(A + (size_t)m0 * CC, (unsigned)(uintptr_t)&Ap[0][0], BM);
        tdm_load_tile(W + (size_t)n0 * CC, (unsigned)(uintptr_t)&Bp[0][0], BN);
    }
#endif

    v8f acc = {};
    for (int step = 0; step < NSTEP; ++step) {
        const int buf = step & 1;
#if HAVE_TDM
        if (wid == 0) {
            if (step + 1 < NSTEP) {
                const int kc = (step + 1) * BK;
                const int nb = (step + 1) & 1;
                tdm_load_tile(A + (size_t)m0 * CC + kc,
                              (unsigned)(uintptr_t)&Ap[nb][0], BM);
                tdm_load_tile(W + (size_t)n0 * CC + kc,
                              (unsigned)(uintptr_t)&Bp[nb][0], BN);
                __builtin_amdgcn_s_wait_tensorcnt(2);  // current chunk done
            } else {
                __builtin_amdgcn_s_wait_tensorcnt(0);
            }
        }
        __syncthreads();               // LDS buffer `buf` ready for all waves
#else
        // Fallback staging: cooperative copy with explicit pad stride.
        {
            const int kc = step * BK;
            for (int i = tid; i < BM * BK; i += 256)
                Ap[buf][(i >> 5) * LDS_STRIDE + (i & 31)] =
                    A[(size_t)(m0 + (i >> 5)) * CC + kc + (i & 31)];
            for (int i = tid; i < BN * BK; i += 256)
                Bp[buf][(i >> 5) * LDS_STRIDE + (i & 31)] =
                    W[(size_t)(n0 + (i >> 5)) * CC + kc + (i & 31)];
        }
        __syncthreads();
#endif
        const float* Ab = &Ap[buf][(wm * 16 + lr) * LDS_STRIDE + 2 * half];
        const float* Bb = &Bp[buf][(wn * 16 + lr) * LDS_STRIDE + 2 * half];
#pragma unroll
        for (int k = 0; k < BK; k += 4) {
            v2f a = *(const v2f*)(Ab + k);
            v2f b = *(const v2f*)(Bb + k);
            acc = __builtin_amdgcn_wmma_f32_16x16x4_f32(
                false, a, false, b, (short)0, acc, false, false);
        }
        __syncthreads();               // done reading before buffer reuse
    }

    const int col = n0 + wn * 16 + lr;
    const float bn = bias ? bias[col] : 0.0f;
#pragma unroll
    for (int i = 0; i < 8; ++i) {
        const int row = m0 + wm * 16 + i + 8 * half;
        out[(size_t)row * CC + col] = silu_f(acc[i] + bn);
    }
}

// ---------------------------------------------------------------------------
// Kernel 3: gather conv. One 64-thread group per (b,l,h); thread = channel d.
// ---------------------------------------------------------------------------
__global__ void gather_conv_kernel(const float* __restrict__ x,
                                   const float* __restrict__ freq,
                                   const float* __restrict__ phase,
                                   const float* __restrict__ kern,
                                   float* __restrict__ hidden) {
    const int gid = blockIdx.x * 4 + (threadIdx.x >> 6);   // (b*L+l)*H + h
    const int d   = threadIdx.x & 63;
    const int h   = gid % HH;
    const int bl  = gid / HH;
    const int l   = bl % LL;
    const int b   = bl / LL;

    const float f = freq[gid];
    const float p = phase[gid];
    const float* kr = kern + (size_t)bl * CC + h * DD;
    const float* xb = x + ((size_t)b * LL) * CC + h * DD + d;

    float acc = 0.0f;
#pragma unroll
    for (int si = 0; si < SS; ++si) {
        const float s  = (float)(si - HALF_S);
        const float sf = s * f;
        const float off = sf + p;
        const int   pi  = l + (int)rintf(off);
        const bool  valid = (pi >= 0) && (pi < LL);
        const int   pc  = min(max(pi, 0), LL - 1);
        float t = (sf + MAX_RECEPTIVE) * (1.0f / (2.0f * MAX_RECEPTIVE)) * (float)(KK - 1);
        t = fminf(fmaxf(t, 0.0f), (float)(KK - 1));
        const float k0f = floorf(t);
        const float w   = t - k0f;
        const int k0 = (int)k0f;
        const int k1 = min(k0 + 1, KK - 1);
        float kv = kr[k0] * (1.0f - w) + kr[k1] * w;
        kv = valid ? kv : 0.0f;
        acc = fmaf(xb[(size_t)pc * CC], kv, acc);
    }
    hidden[(size_t)bl * CC + h * DD + d] = acc;
}

// ---------------------------------------------------------------------------
// Launch
// ---------------------------------------------------------------------------
extern "C" void kernel_launch(void* const* d_in, const int* in_sizes, int n_in,
                              void* d_out, int out_size, void* d_ws, size_t ws_size,
                              hipStream_t stream) {
    const float* x  = (const float*)d_in[0];
    const float* Ww = (const float*)d_in[1];
    const float* bw = (const float*)d_in[2];
    const float* Wk = (const float*)d_in[3];
    const float* bk = (const float*)d_in[4];
    const float* Wo = (const float*)d_in[5];
    float* out = (float*)d_out;

    float* ws     = (float*)d_ws;
    float* freq   = ws;
    float* phase  = freq + (size_t)MROWS * HH;
    float* kern   = phase + (size_t)MROWS * HH;
    float* hidden = kern + (size_t)MROWS * CC;

    // 1) wave -> freq/phase
    wave_kernel<<<MROWS / 16, 256, 0, stream>>>(x, Ww, bw, freq, phase);

    // 2) kernel values = silu(x @ Wk^T + bk)   [4096, 512]
    gemm_tdm_wmma_silu<<<dim3(CC / BN, MROWS / BM), 256, 0, stream>>>(
        x, Wk, bk, kern);

    // 3) gather conv -> hidden
    gather_conv_kernel<<<(MROWS * HH) / 4, 256, 0, stream>>>(x, freq, phase,
                                                             kern, hidden);

    // 4) out = silu(hidden @ Wo^T)   [4096, 512]
    gemm_tdm_wmma_silu<<<dim3(CC / BN, MROWS / BM), 256, 0, stream>>>(
        hidden, Wo, nullptr, out);
}